// Coords2Stress_17231408791693
// MI455X (gfx1250) — compile-verified
//
#include <hip/hip_runtime.h>
#include <hip/hip_bf16.h>

typedef __attribute__((ext_vector_type(2))) float v2f;
typedef __attribute__((ext_vector_type(4))) float v4f;
typedef __attribute__((ext_vector_type(8))) float v8f;

#define EPS_F 1e-5f

// Exact 32-lane sum using V_WMMA_F32_16X16X4_F32 (documented A and C/D layouts).
// A = {v, 0}: row m of A = [v_m, 0, v_{m+16}, 0]; B = all ones (layout-proof)
// -> D[m,n] = v_m + v_{m+16}. Lanes 0-15 hold rows 0..7 in the 8 D VGPRs,
// lanes 16-31 rows 8..15, so local 8-sum + shfl_xor(16) = full 32-lane sum.
__device__ __forceinline__ float wave_sum_wmma(float v) {
    v2f a;  a[0]  = v;    a[1]  = 0.0f;
    v2f bo; bo[0] = 1.0f; bo[1] = 1.0f;
    v8f c = {};
    c = __builtin_amdgcn_wmma_f32_16x16x4_f32(false, a, false, bo,
                                              (short)0, c, false, false);
    float s = ((c[0] + c[1]) + (c[2] + c[3])) + ((c[4] + c[5]) + (c[6] + c[7]));
    s += __shfl_xor(s, 16, 32);
    return s;
}

// One 64-thread block (2 waves) per (atom row i, batch b).
// Each thread owns 16 j's as 4 groups of 4 consecutive j (j = 4*tid + 256*g),
// writes 3 output rows with aligned NT b128 stores, fully coalesced.
__global__ __launch_bounds__(64) void
coords2hess_kernel(const float* __restrict__ coords,
                   const int* __restrict__ num_atoms,
                   float* __restrict__ out, int N) {
    const int i   = blockIdx.x;
    const int b   = blockIdx.y;
    const int tid = threadIdx.x;

    __shared__ v4f   sc4[768];      // 1024 atoms * 3 floats = 12 KB
    __shared__ float wsum[2][6];    // per-wave partial diagonal sums

    const v4f* gp = reinterpret_cast<const v4f*>(coords + (size_t)b * 3 * N);
#pragma unroll
    for (int r = 0; r < 12; ++r) sc4[tid + 64 * r] = gp[tid + 64 * r];
    const int na = num_atoms[b];
    __syncthreads();

    const float* sc  = reinterpret_cast<const float*>(sc4);
    const float  cix = sc[3 * i + 0];
    const float  ciy = sc[3 * i + 1];
    const float  ciz = sc[3 * i + 2];
    const int    jlim = (i < na) ? na : 0;   // fold row validity into pair mask

    const size_t ld = (size_t)3 * N;
    float* const o0 = out + ((size_t)b * 3 * N + (size_t)3 * i) * ld;
    float* const o1 = o0 + ld;
    float* const o2 = o1 + ld;

    float a0 = 0.f, a1 = 0.f, a2 = 0.f, a3 = 0.f, a4 = 0.f, a5 = 0.f;

#pragma unroll
    for (int g = 0; g < 4; ++g) {
        const int j0  = 4 * tid + 256 * g;
        const int v4i = 3 * tid + 192 * g;   // 48B/thread-group: v4f index in a
                                             // 3N-float row (LDS and output alike)
        const v4f c0 = sc4[v4i + 0];
        const v4f c1 = sc4[v4i + 1];
        const v4f c2 = sc4[v4i + 2];
        const float f[12] = {c0[0], c0[1], c0[2], c0[3],
                             c1[0], c1[1], c1[2], c1[3],
                             c2[0], c2[1], c2[2], c2[3]};
        float r0b[12], r1b[12], r2b[12];
#pragma unroll
        for (int jj = 0; jj < 4; ++jj) {
            const float dx = cix - f[3 * jj + 0];
            const float dy = ciy - f[3 * jj + 1];
            const float dz = ciz - f[3 * jj + 2];
            const float d2 = fmaf(dx, dx, fmaf(dy, dy, fmaf(dz, dz, EPS_F)));
            const float rc = __builtin_amdgcn_rcpf(d2);
            const float q  = ((j0 + jj) < jlim) ? rc : 0.0f;  // mask folded in
            const float gx = dx * q, gy = dy * q, gz = dz * q;
            // 6 unique symmetric entries; neg folds into the mul
            const float n00 = -(gx * dx);
            const float n01 = -(gx * dy);
            const float n02 = -(gx * dz);
            const float n11 = -(gy * dy);
            const float n12 = -(gy * dz);
            const float n22 = -(gz * dz);
            a0 -= n00; a1 -= n01; a2 -= n02; a3 -= n11; a4 -= n12; a5 -= n22;
            r0b[3*jj+0] = n00; r0b[3*jj+1] = n01; r0b[3*jj+2] = n02;
            r1b[3*jj+0] = n01; r1b[3*jj+1] = n11; r1b[3*jj+2] = n12;
            r2b[3*jj+0] = n02; r2b[3*jj+1] = n12; r2b[3*jj+2] = n22;
        }
#pragma unroll
        for (int k = 0; k < 3; ++k) {
            v4f s0 = {r0b[4*k], r0b[4*k+1], r0b[4*k+2], r0b[4*k+3]};
            v4f s1 = {r1b[4*k], r1b[4*k+1], r1b[4*k+2], r1b[4*k+3]};
            v4f s2 = {r2b[4*k], r2b[4*k+1], r2b[4*k+2], r2b[4*k+3]};
            __builtin_nontemporal_store(s0, reinterpret_cast<v4f*>(o0) + v4i + k);
            __builtin_nontemporal_store(s1, reinterpret_cast<v4f*>(o1) + v4i + k);
            __builtin_nontemporal_store(s2, reinterpret_cast<v4f*>(o2) + v4i + k);
        }
    }

    // Diagonal block = +sum_j sep⊗sep/d2: WMMA wave reduction (EXEC all ones,
    // uniform control flow here), then 2-wave combine through LDS.
    const float s0 = wave_sum_wmma(a0);
    const float s1 = wave_sum_wmma(a1);
    const float s2 = wave_sum_wmma(a2);
    const float s3 = wave_sum_wmma(a3);
    const float s4 = wave_sum_wmma(a4);
    const float s5 = wave_sum_wmma(a5);
    const int lane = tid & 31, wv = tid >> 5;
    if (lane == 0) {
        wsum[wv][0] = s0; wsum[wv][1] = s1; wsum[wv][2] = s2;
        wsum[wv][3] = s3; wsum[wv][4] = s4; wsum[wv][5] = s5;
    }
    __syncthreads();

    // The unique thread that owns j==i (it stored the zero block there in the
    // main pass) patches the 3x3 diagonal block. Same-wave stores to the same
    // address complete in order (ISA: stores are ordered with stores), so no
    // cross-thread WAW hazard and zero cost in the hot path.
    if (tid == ((i & 255) >> 2)) {
        float d[6];
#pragma unroll
        for (int c = 0; c < 6; ++c) d[c] = wsum[0][c] + wsum[1][c];
        o0[3*i+0] = d[0]; o0[3*i+1] = d[1]; o0[3*i+2] = d[2];
        o1[3*i+0] = d[1]; o1[3*i+1] = d[3]; o1[3*i+2] = d[4];
        o2[3*i+0] = d[2]; o2[3*i+1] = d[4]; o2[3*i+2] = d[5];
    }
}

extern "C" void kernel_launch(void* const* d_in, const int* in_sizes, int n_in,
                              void* d_out, int out_size, void* d_ws, size_t ws_size,
                              hipStream_t stream) {
    (void)n_in; (void)out_size; (void)d_ws; (void)ws_size;
    const float* coords    = (const float*)d_in[0];
    const int*   num_atoms = (const int*)d_in[1];
    float*       out       = (float*)d_out;

    const int B = in_sizes[1];               // 8
    const int N = in_sizes[0] / (3 * B);     // 1024

    dim3 grid(N, B);                         // one block per (i, b)
    coords2hess_kernel<<<grid, 64, 0, stream>>>(coords, num_atoms, out, N);
}